// Simple_ST_GNN_70489003261973
// MI455X (gfx1250) — compile-verified
//
#include <hip/hip_runtime.h>
#include <hip/hip_bf16.h>

typedef float v2f __attribute__((ext_vector_type(2)));
typedef float v8f __attribute__((ext_vector_type(8)));

#define N_NODES 100000
#define F_IN    128
#define HDIM    64
#define MTILES  (N_NODES / 16)   // 6250
#define NTILES  (HDIM / 16)      // 4

__device__ __forceinline__ float sigmoidf_(float x) {
    return 1.0f / (1.0f + __expf(-x));
}

// ---------------- degree / normalization ----------------

__global__ void k_deg_init(float* __restrict__ deg, int n) {
    int i = blockIdx.x * blockDim.x + threadIdx.x;
    if (i < n) deg[i] = 1.0f;                       // self-loop contribution
}

__global__ void k_deg_edges(const long long* __restrict__ col,
                            float* __restrict__ deg, long long E) {
    long long e = (long long)blockIdx.x * blockDim.x + threadIdx.x;
    if (e < E) atomicAdd(&deg[(int)col[e]], 1.0f);
}

__global__ void k_dis(float* __restrict__ deg, int n) {
    int i = blockIdx.x * blockDim.x + threadIdx.x;
    if (i < n) {
        float d = deg[i];
        deg[i] = (d > 0.0f) ? rsqrtf(d) : 0.0f;     // in place: deg -> dis
    }
}

// ------- GEMM1: xt = x @ W_gcn (f32 WMMA 16x16x4, 4 accumulators/wave) -------
// One wave owns a full 16-row x 64-col output slab: the A fragment (the HBM-
// heavy operand) is loaded once per k-step and reused across 4 WMMAs.

__global__ __launch_bounds__(256)
void k_gcn_gemm(const float* __restrict__ x, const float* __restrict__ Wg,
                float* __restrict__ xt) {
    const int wave = blockIdx.x * (blockDim.x >> 5) + (threadIdx.x >> 5);
    const int lane = threadIdx.x & 31;
    if (wave >= MTILES) return;                     // wave-uniform: whole wave exits

    const int m0   = wave * 16;
    const int l16  = lane & 15;                     // row for A, col for B/C/D
    const int half = lane >> 4;                     // K-half selector

    v8f acc[NTILES] = {};
    const float* arow = x + (size_t)(m0 + l16) * F_IN;
    for (int k0 = 0; k0 < F_IN; k0 += 4) {
        const int kk = k0 + half * 2;
        v2f a = *(const v2f*)(arow + kk);           // A[m][kk], A[m][kk+1]
        const float* w0 = Wg + (size_t)kk * HDIM + l16;   // row kk of W
        const float* w1 = w0 + HDIM;                      // row kk+1 of W
#pragma unroll
        for (int nt = 0; nt < NTILES; ++nt) {
            v2f b;
            b.x = w0[nt * 16];
            b.y = w1[nt * 16];
            acc[nt] = __builtin_amdgcn_wmma_f32_16x16x4_f32(false, a, false, b,
                                                  (short)0, acc[nt], false, false);
        }
    }
    // C/D layout: VGPR v, lane -> row m0 + v + 8*half, col nt*16 + (lane&15)
#pragma unroll
    for (int nt = 0; nt < NTILES; ++nt) {
        const int coln = nt * 16 + l16;
#pragma unroll
        for (int v = 0; v < 8; ++v)
            xt[(size_t)(m0 + v + 8 * half) * HDIM + coln] = acc[nt][v];
    }
}

// ---------------- self-loop + bias init of aggregate ----------------

__global__ void k_agg_init(const float* __restrict__ dis, const float* __restrict__ xt,
                           const float* __restrict__ bg, float* __restrict__ agg,
                           int total) {
    int i = blockIdx.x * blockDim.x + threadIdx.x;
    if (i < total) {
        int node = i >> 6;          // / HDIM
        int h    = i & 63;
        float d  = dis[node];
        agg[i]   = d * d * xt[i] + bg[h];
    }
}

// ---------------- edge scatter-add (16 threads/edge, L2-resident atomics) ----

__global__ __launch_bounds__(256)
void k_agg_edges(const long long* __restrict__ rows, const long long* __restrict__ cols,
                 const float* __restrict__ dis, const float* __restrict__ xt,
                 float* __restrict__ agg, long long E) {
    long long t = (long long)blockIdx.x * blockDim.x + threadIdx.x;
    long long e = t >> 4;
    int q = (int)(t & 15);                          // which float4 of the 64-wide row
    if (e >= E) return;
    if (q == 0 && e + 2048 < E) {
        __builtin_prefetch(&rows[e + 2048], 0, 0);  // global_prefetch_b8 on index stream
        __builtin_prefetch(&cols[e + 2048], 0, 0);
    }
    int r = (int)rows[e];
    int c = (int)cols[e];
    float w = dis[r] * dis[c];
    const float4 v = *(const float4*)(xt + (size_t)r * HDIM + q * 4);
    float* dst = agg + (size_t)c * HDIM + q * 4;
    atomicAdd(dst + 0, w * v.x);
    atomicAdd(dst + 1, w * v.y);
    atomicAdd(dst + 2, w * v.z);
    atomicAdd(dst + 3, w * v.w);
}

// ---------------- fused GRU step (6 WMMA accumulators per wave) ----------------

__global__ __launch_bounds__(256)
void k_gru(const float* __restrict__ agg, const float* __restrict__ h_in,
           const float* __restrict__ Wih, const float* __restrict__ Whh,
           const float* __restrict__ bih, const float* __restrict__ bhh,
           float* __restrict__ out) {
    const int wave = blockIdx.x * (blockDim.x >> 5) + (threadIdx.x >> 5);
    const int lane = threadIdx.x & 31;
    const int mt = wave / NTILES;
    const int ht = wave % NTILES;
    if (mt >= MTILES) return;

    const int m0   = mt * 16;
    const int h0   = ht * 16;
    const int l16  = lane & 15;
    const int half = lane >> 4;

    // acc[0..2] = gi gates (r,z,n), acc[3..5] = gh gates (r,z,n)
    v8f acc[6] = {};
    const float* ga = agg  + (size_t)(m0 + l16) * HDIM;   // A row: relu(gcn out)
    const float* hh = h_in + (size_t)(m0 + l16) * HDIM;   // A row: prev hidden

    for (int k0 = 0; k0 < HDIM; k0 += 4) {
        const int kk = k0 + half * 2;
        v2f ax = *(const v2f*)(ga + kk);
        ax.x = fmaxf(ax.x, 0.0f);                         // fused relu
        ax.y = fmaxf(ax.y, 0.0f);
        v2f ah = *(const v2f*)(hh + kk);
#pragma unroll
        for (int g = 0; g < 3; ++g) {
            const int col = g * HDIM + h0 + l16;          // row of W (=col of W^T)
            v2f bi = *(const v2f*)(Wih + (size_t)col * HDIM + kk);  // contiguous
            v2f bh = *(const v2f*)(Whh + (size_t)col * HDIM + kk);
            acc[g]     = __builtin_amdgcn_wmma_f32_16x16x4_f32(false, ax, false, bi,
                                                (short)0, acc[g],     false, false);
            acc[3 + g] = __builtin_amdgcn_wmma_f32_16x16x4_f32(false, ah, false, bh,
                                                (short)0, acc[3 + g], false, false);
        }
    }

    const int coln = h0 + l16;
    const float bir = bih[coln],            bhr = bhh[coln];
    const float biz = bih[HDIM + coln],     bhz = bhh[HDIM + coln];
    const float bin = bih[2 * HDIM + coln], bhn = bhh[2 * HDIM + coln];

#pragma unroll
    for (int v = 0; v < 8; ++v) {
        const int row = m0 + v + 8 * half;
        float i_r = acc[0][v] + bir, h_r = acc[3][v] + bhr;
        float i_z = acc[1][v] + biz, h_z = acc[4][v] + bhz;
        float i_n = acc[2][v] + bin, h_n = acc[5][v] + bhn;
        float r  = sigmoidf_(i_r + h_r);
        float z  = sigmoidf_(i_z + h_z);
        float nn = tanhf(i_n + r * h_n);
        float hp = h_in[(size_t)row * HDIM + coln];
        float hn_out = (1.0f - z) * nn + z * hp;
        size_t idx = (size_t)row * HDIM + coln;
        out[idx] = hn_out;                               // output 0: h_new
        out[(size_t)N_NODES * HDIM + idx] = hn_out;      // output 1: new hidden (same)
    }
}

// ---------------- launcher ----------------

extern "C" void kernel_launch(void* const* d_in, const int* in_sizes, int n_in,
                              void* d_out, int out_size, void* d_ws, size_t ws_size,
                              hipStream_t stream) {
    const float*     x   = (const float*)d_in[0];
    const long long* ei  = (const long long*)d_in[1];   // int64 (2, E)
    const float*     hid = (const float*)d_in[2];       // (1, N, H) flat
    const float*     Wg  = (const float*)d_in[3];
    const float*     bg  = (const float*)d_in[4];
    const float*     Wih = (const float*)d_in[5];
    const float*     Whh = (const float*)d_in[6];
    const float*     bih = (const float*)d_in[7];
    const float*     bhh = (const float*)d_in[8];
    float*           out = (float*)d_out;

    const long long E = (long long)(in_sizes[1] / 2);
    const long long* rows = ei;          // edge_index[0] = source
    const long long* cols = ei + E;      // edge_index[1] = target

    float* ws  = (float*)d_ws;
    float* xt  = ws;                                 // N*H
    float* agg = ws + (size_t)N_NODES * HDIM;        // N*H
    float* dis = agg + (size_t)N_NODES * HDIM;       // N (deg, then rsqrt in place)

    // 1-3: symmetric normalization coefficients
    k_deg_init <<<(N_NODES + 255) / 256, 256, 0, stream>>>(dis, N_NODES);
    k_deg_edges<<<(int)((E + 255) / 256), 256, 0, stream>>>(cols, dis, E);
    k_dis      <<<(N_NODES + 255) / 256, 256, 0, stream>>>(dis, N_NODES);

    // 4: xt = x @ W_gcn via f32 WMMA (6250 waves, 8 waves/block, 4 acc/wave)
    k_gcn_gemm<<<(MTILES + 7) / 8, 256, 0, stream>>>(x, Wg, xt);

    // 5: agg = self-loop term + bias
    k_agg_init<<<(N_NODES * HDIM + 255) / 256, 256, 0, stream>>>(dis, xt, bg, agg,
                                                                 N_NODES * HDIM);
    // 6: edge scatter-add (16 threads per edge)
    long long ethreads = E * 16;
    k_agg_edges<<<(int)((ethreads + 255) / 256), 256, 0, stream>>>(rows, cols, dis,
                                                                   xt, agg, E);
    // 7: fused relu + GRU step via f32 WMMA, writes both tuple outputs
    const int gru_waves = MTILES * NTILES;
    k_gru<<<(gru_waves + 7) / 8, 256, 0, stream>>>(agg, hid, Wih, Whh, bih, bhh, out);
}